// TemporalEncoderMambaPooling_80891414052975
// MI455X (gfx1250) — compile-verified
//
#include <hip/hip_runtime.h>
#include <math.h>

typedef __attribute__((ext_vector_type(2))) float v2f;
typedef __attribute__((ext_vector_type(8))) float v8f;
typedef __attribute__((ext_vector_type(4))) unsigned int v4u;
typedef __attribute__((ext_vector_type(8))) int v8i;
typedef __attribute__((ext_vector_type(4))) int v4i;

#ifndef __has_builtin
#define __has_builtin(x) 0
#endif
#if __has_builtin(__builtin_amdgcn_tensor_load_to_lds) && __has_builtin(__builtin_amdgcn_s_wait_tensorcnt)
#define USE_TDM 1
#else
#define USE_TDM 0
#endif

#define TOK   16384   // B*L tokens
#define SEQ   4096
#define NBATCH 4
#define DM    256
#define DI    512
#define XDW   48      // DT_RANK(16) + 2*D_STATE(32)

#define BM 64
#define BN 64
#define BKK 16

#if USE_TDM
// Issue a 2D tile load via the Tensor Data Mover (D# per CDNA5 ISA).
// dims/strides in elements (data_size units), addresses in bytes.
// This toolchain exposes the 6-arg builtin:
//   (uint32x4 g0, int32x8 g1, int32x4 g2, int32x4 g3, int32x8 gx, i32 cpol)
__device__ __forceinline__ void tdm_load_2d(const float* gptr, unsigned lds_off,
                                            unsigned tensor_d0, unsigned tensor_d1,
                                            unsigned tile_d0, unsigned tile_d1,
                                            unsigned stride0, unsigned g1d0)
{
  unsigned long long ga = (unsigned long long)(uintptr_t)gptr;
  v4u g0;
  g0.x = 1u;                                              // count=1 (valid, user mode)
  g0.y = lds_off;                                         // lds_addr [63:32]
  g0.z = (unsigned)(ga & 0xffffffffu);                    // global_addr [95:64]
  g0.w = (unsigned)((ga >> 32) & 0x1ffffffu) | (2u << 30);// addr[56:32] | type=2
  v8i g1;
  g1[0] = (int)g1d0;                                      // data_size=4B (+pad cfg)
  g1[1] = (int)((tensor_d0 & 0xffffu) << 16);             // tensor_dim0 lo16 @ [63:48]
  g1[2] = (int)(((tensor_d0 >> 16) & 0xffffu) | ((tensor_d1 & 0xffffu) << 16));
  g1[3] = (int)(((tensor_d1 >> 16) & 0xffffu) | ((tile_d0 & 0xffffu) << 16));
  g1[4] = (int)(tile_d1 & 0xffffu);                       // tile_dim1; tile_dim2=0
  g1[5] = (int)stride0;                                   // tensor_dim0_stride lo32
  g1[6] = 0;                                              // stride hi / dim1_stride
  g1[7] = 0;
  v4i z4 = {0, 0, 0, 0};
  v8i z8 = {0, 0, 0, 0, 0, 0, 0, 0};
  __builtin_amdgcn_tensor_load_to_lds(g0, g1, z4, z4, z8, 0);
}
#endif

// ================= WMMA f32 GEMM (TDM-fed, double buffered) =================
// C[M,N] = A[M,K] @ B[K,N] ; A row stride = lda. epi==1: softplus(C + bias[n])
__global__ void __launch_bounds__(256)
gemm_wmma_f32(const float* __restrict__ A, int lda,
              const float* __restrict__ Bw,
              float* __restrict__ C,
              int M, int N, int K,
              const float* __restrict__ bias, int epi)
{
  __shared__ float As[2][BM * 17];   // 16 data dwords + 1 pad dword per row
  __shared__ float Bs[2][BKK * BN];
  const int tid  = threadIdx.x;
  const int n0   = blockIdx.x * BN;
  const int m0   = blockIdx.y * BM;
  const int wave = tid >> 5;
  const int lane = tid & 31;
  const int hl   = (lane >> 4) & 1;   // K-pair select (lanes 16-31 take K+2,K+3)
  const int lm   = lane & 15;
  const int wm   = wave & 3;          // 4 wave-tiles along M
  const int wn   = wave >> 2;         // 2 wave-tiles along N (32 wide each)

  v8f acc0 = {}; v8f acc1 = {};

#if USE_TDM
  // A tile: 64 rows x 16 cols, pad 1 dword per 16 (interval code 3, amount 0)
  const unsigned A_FLAGS = 0x00D20000u;  // data_size=2, pad_en, interval=16dw
  const unsigned B_FLAGS = 0x00020000u;  // data_size=2
  auto issue = [&](int kq, int buf) {
    tdm_load_2d(A + (size_t)m0 * lda + kq,
                (unsigned)(uintptr_t)&As[buf][0],
                (unsigned)(K - kq), (unsigned)(M - m0),
                BKK, BM, (unsigned)lda, A_FLAGS);
    tdm_load_2d(Bw + (size_t)kq * N + n0,
                (unsigned)(uintptr_t)&Bs[buf][0],
                (unsigned)(N - n0), (unsigned)(K - kq),
                BN, BKK, (unsigned)N, B_FLAGS);
  };
  if (wave == 0) issue(0, 0);
  for (int k0 = 0; k0 < K; k0 += BKK) {
    const int buf = (k0 >> 4) & 1;
    if (wave == 0) __builtin_amdgcn_s_wait_tensorcnt(0);  // chunk in flight done
    __syncthreads();   // all waves: chunk visible; prev buffer readers retired
    if (wave == 0 && k0 + BKK < K) issue(k0 + BKK, buf ^ 1);
#pragma unroll
    for (int kk = 0; kk < BKK; kk += 4) {
      v2f a, b0, b1;
      const float* ar = &As[buf][(wm*16 + lm) * 17 + kk + 2*hl];
      a.x = ar[0]; a.y = ar[1];
      const float* br = &Bs[buf][(kk + 2*hl) * BN + wn*32 + lm];
      b0.x = br[0];  b0.y = br[BN];
      b1.x = br[16]; b1.y = br[16 + BN];
      acc0 = __builtin_amdgcn_wmma_f32_16x16x4_f32(false, a, false, b0, (short)0, acc0, false, false);
      acc1 = __builtin_amdgcn_wmma_f32_16x16x4_f32(false, a, false, b1, (short)0, acc1, false, false);
    }
  }
#else
  // Fallback: VGPR-staged LDS tiles
  const int arow = tid >> 2, aq = (tid & 3) << 2;
  const int brow = tid >> 4, bq = (tid & 15) << 2;
  for (int k0 = 0; k0 < K; k0 += BKK) {
    const float* ap = A + (size_t)(m0 + arow) * lda + (k0 + aq);
    float4 av = *(const float4*)ap;
    if (k0 + BKK < K) __builtin_prefetch(ap + BKK, 0, 1);
    float* asp = &As[0][arow * 17 + aq];
    asp[0] = av.x; asp[1] = av.y; asp[2] = av.z; asp[3] = av.w;
    const int gn = n0 + bq;
    const float* bp = Bw + (size_t)(k0 + brow) * N + gn;
    float4 bv;
    if (gn + 3 < N) bv = *(const float4*)bp;
    else {
      bv.x = (gn + 0 < N) ? bp[0] : 0.f;
      bv.y = (gn + 1 < N) ? bp[1] : 0.f;
      bv.z = (gn + 2 < N) ? bp[2] : 0.f;
      bv.w = (gn + 3 < N) ? bp[3] : 0.f;
    }
    float* bsp = &Bs[0][brow * BN + bq];
    bsp[0]=bv.x; bsp[1]=bv.y; bsp[2]=bv.z; bsp[3]=bv.w;
    __syncthreads();
#pragma unroll
    for (int kk = 0; kk < BKK; kk += 4) {
      v2f a, b0, b1;
      const float* ar = &As[0][(wm*16 + lm) * 17 + kk + 2*hl];
      a.x = ar[0]; a.y = ar[1];
      const float* br = &Bs[0][(kk + 2*hl) * BN + wn*32 + lm];
      b0.x = br[0];  b0.y = br[BN];
      b1.x = br[16]; b1.y = br[16 + BN];
      acc0 = __builtin_amdgcn_wmma_f32_16x16x4_f32(false, a, false, b0, (short)0, acc0, false, false);
      acc1 = __builtin_amdgcn_wmma_f32_16x16x4_f32(false, a, false, b1, (short)0, acc1, false, false);
    }
    __syncthreads();
  }
#endif

  // C/D layout: VGPR r -> M = r (lanes0-15) / 8+r (lanes16-31), N = lane&15
  const int mb  = m0 + wm*16 + hl*8;
  const int nc0 = n0 + wn*32 + lm;
  const int nc1 = nc0 + 16;
#pragma unroll
  for (int r = 0; r < 8; ++r) {
    const size_t m = (size_t)(mb + r);
    float v0 = acc0[r], v1 = acc1[r];
    if (nc0 < N) {
      if (epi) { float t = v0 + bias[nc0]; v0 = (t > 20.f) ? t : __logf(1.f + __expf(t)); }
      C[m * (size_t)N + nc0] = v0;
    }
    if (nc1 < N) {
      if (epi) { float t = v1 + bias[nc1]; v1 = (t > 20.f) ? t : __logf(1.f + __expf(t)); }
      C[m * (size_t)N + nc1] = v1;
    }
  }
}

// ================= in_proj + flip =================
__global__ void __launch_bounds__(256)
inproj_flip(const float* __restrict__ y, const float* __restrict__ w,
            const float* __restrict__ b, float* __restrict__ x, float* __restrict__ xr)
{
  int idx = blockIdx.x * 256 + threadIdx.x;    // TOK*DM
  int d = idx & (DM - 1), row = idx >> 8;
  float v = y[row] * w[d] + b[d];
  x[idx] = v;
  int l = row & (SEQ - 1), bb = row >> 12;
  xr[(size_t)((bb << 12) + (SEQ - 1 - l)) * DM + d] = v;
}

// ================= depthwise causal conv(4) + SiLU =================
__global__ void __launch_bounds__(256)
conv_silu(const float* __restrict__ xz, const float* __restrict__ cw,
          const float* __restrict__ cb, float* __restrict__ xc)
{
  int idx = blockIdx.x * 256 + threadIdx.x;    // TOK*DI
  int d = idx & (DI - 1);
  int row = idx >> 9;
  int l = row & (SEQ - 1);
  int r0 = row - l;
  float acc = cb[d];
#pragma unroll
  for (int k = 0; k < 4; ++k) {
    int ll = l - 3 + k;
    if (ll >= 0) acc += xz[(size_t)(r0 + ll) * 1024 + d] * cw[k * DI + d];
  }
  xc[(size_t)row * DI + d] = acc / (1.f + __expf(-acc));
}

// ================= selective scan (fused D-skip + SiLU gate) =================
__global__ void __launch_bounds__(256)
scan_fused(const float* __restrict__ dt, const float* __restrict__ xc,
           const float* __restrict__ xz,    // z at col 512+d (row width 1024)
           const float* __restrict__ xdbl,  // row width 48: B at 16+n, C at 32+n
           const float* __restrict__ A_log, const float* __restrict__ Dp,
           float* __restrict__ yv)          // may alias dt
{
  __shared__ float sBC[32];
  const int tid = threadIdx.x;
  const int batch = blockIdx.x >> 1;
  const int d = ((blockIdx.x & 1) << 8) + tid;
  float Ar[16], h[16];
#pragma unroll
  for (int n = 0; n < 16; ++n) { Ar[n] = -__expf(A_log[d * 16 + n]); h[n] = 0.f; }
  const float Dd = Dp[d];
  const size_t rowbase = (size_t)batch * SEQ;
  for (int l = 0; l < SEQ; ++l) {
    const size_t row = rowbase + l;
    if (tid < 32) sBC[tid] = xdbl[row * XDW + 16 + tid];
    __syncthreads();
    const float dtv = dt[row * DI + d];
    const float u   = xc[row * DI + d];
    const float zv  = xz[row * 1024 + DI + d];
    const float dtu = dtv * u;
    float y = 0.f;
#pragma unroll
    for (int n = 0; n < 16; ++n) {
      float dA = __expf(dtv * Ar[n]);
      h[n] = h[n] * dA + dtu * sBC[n];
      y += h[n] * sBC[16 + n];
    }
    float o = y + u * Dd;
    float sg = zv / (1.f + __expf(-zv));
    yv[row * DI + d] = o * sg;
    __syncthreads();
  }
}

// ================= LayerNorm (wave per row of 256) =================
__global__ void __launch_bounds__(256)
layernorm(const float* __restrict__ in, const float* __restrict__ g,
          const float* __restrict__ b, float* __restrict__ out)
{
  const int tid = threadIdx.x, w = tid >> 5, lane = tid & 31;
  const size_t row = (size_t)blockIdx.x * 8 + w;
  const float* p = in + row * DM;
  float v[8]; float s = 0.f;
#pragma unroll
  for (int j = 0; j < 8; ++j) { v[j] = p[lane + 32 * j]; s += v[j]; }
#pragma unroll
  for (int o = 16; o > 0; o >>= 1) s += __shfl_xor(s, o, 32);
  const float mean = s * (1.f / 256.f);
  float q = 0.f;
#pragma unroll
  for (int j = 0; j < 8; ++j) { float dd = v[j] - mean; q += dd * dd; }
#pragma unroll
  for (int o = 16; o > 0; o >>= 1) q += __shfl_xor(q, o, 32);
  const float rs = rsqrtf(q * (1.f / 256.f) + 1e-5f);
#pragma unroll
  for (int j = 0; j < 8; ++j) {
    int c = lane + 32 * j;
    out[row * DM + c] = (v[j] - mean) * rs * g[c] + b[c];
  }
}

// ================= attention logits (wave per row) =================
__global__ void __launch_bounds__(256)
attn_logits(const float* __restrict__ h, const float* __restrict__ aw,
            float* __restrict__ lg)
{
  const int tid = threadIdx.x, w = tid >> 5, lane = tid & 31;
  const size_t row = (size_t)blockIdx.x * 8 + w;
  const float* p = h + row * DM;
  float s = 0.f;
#pragma unroll
  for (int j = 0; j < 8; ++j) { int c = lane + 32 * j; s += p[c] * aw[c]; }
#pragma unroll
  for (int o = 16; o > 0; o >>= 1) s += __shfl_xor(s, o, 32);
  if (lane == 0) lg[row] = s;
}

// ================= softmax pool + last-token extract =================
__global__ void __launch_bounds__(256)
pool_softmax(const float* __restrict__ h, const float* __restrict__ logits,
             float* __restrict__ zp, float* __restrict__ zl, int off)
{
  __shared__ float red[256];
  const int tid = threadIdx.x, b = blockIdx.x;
  const float* lg = logits + (size_t)b * SEQ;
  float mx = -3.4e38f;
  for (int t = tid; t < SEQ; t += 256) mx = fmaxf(mx, lg[t]);
  red[tid] = mx; __syncthreads();
  for (int s = 128; s > 0; s >>= 1) { if (tid < s) red[tid] = fmaxf(red[tid], red[tid + s]); __syncthreads(); }
  mx = red[0]; __syncthreads();
  float se = 0.f;
  for (int t = tid; t < SEQ; t += 256) se += __expf(lg[t] - mx);
  red[tid] = se; __syncthreads();
  for (int s = 128; s > 0; s >>= 1) { if (tid < s) red[tid] += red[tid + s]; __syncthreads(); }
  const float inv = 1.f / red[0];
  const float* hb = h + (size_t)b * SEQ * DM;
  float acc = 0.f;
  for (int t = 0; t < SEQ; ++t)
    acc += hb[(size_t)t * DM + tid] * (__expf(lg[t] - mx) * inv);
  zp[b * 512 + off + tid] = acc;
  zl[b * 512 + off + tid] = hb[(size_t)(SEQ - 1) * DM + tid];
}

// ================= output MLP head: gelu(z@W1+b1)@W2+b2 =================
__global__ void __launch_bounds__(512)
head_mlp(const float* __restrict__ zin, const float* __restrict__ W1,
         const float* __restrict__ b1, const float* __restrict__ W2,
         const float* __restrict__ b2, float* __restrict__ out)
{
  __shared__ float tmp[512];
  const int b = blockIdx.x, j = threadIdx.x;
  const float* zr = zin + b * 512;
  float s = b1[j];
  for (int k = 0; k < 512; ++k) s += zr[k] * W1[k * 512 + j];
  s = 0.5f * s * (1.f + erff(s * 0.70710678118654752f));   // exact GELU
  tmp[j] = s;
  __syncthreads();
  if (j < 128) {
    float o = b2[j];
    for (int k = 0; k < 512; ++k) o += tmp[k] * W2[k * 128 + j];
    out[b * 128 + j] = o;
  }
}

// ================= host-side orchestration =================
extern "C" void kernel_launch(void* const* d_in, const int* in_sizes, int n_in,
                              void* d_out, int out_size, void* d_ws, size_t ws_size,
                              hipStream_t stream) {
  (void)in_sizes; (void)n_in; (void)out_size; (void)ws_size;
  const float* y   = (const float*)d_in[0];
  const float* ipw = (const float*)d_in[1];
  const float* ipb = (const float*)d_in[2];
  struct Blk { const float *in_w,*conv_w,*conv_b,*x_w,*dt_w,*dt_b,*A_log,*D,*out_w; };
  Blk blk[2];
  for (int i = 0; i < 2; ++i) {
    int base = 3 + 9 * i;
    blk[i].in_w   = (const float*)d_in[base + 0];
    blk[i].conv_w = (const float*)d_in[base + 1];
    blk[i].conv_b = (const float*)d_in[base + 2];
    blk[i].x_w    = (const float*)d_in[base + 3];
    blk[i].dt_w   = (const float*)d_in[base + 4];
    blk[i].dt_b   = (const float*)d_in[base + 5];
    blk[i].A_log  = (const float*)d_in[base + 6];
    blk[i].D      = (const float*)d_in[base + 7];
    blk[i].out_w  = (const float*)d_in[base + 8];
  }
  const float* ln_g = (const float*)d_in[21];
  const float* ln_b = (const float*)d_in[22];
  const float* attw = (const float*)d_in[23];
  const float* g1w  = (const float*)d_in[24];
  const float* g1b  = (const float*)d_in[25];
  const float* g2w  = (const float*)d_in[26];
  const float* g2b  = (const float*)d_in[27];
  const float* t1w  = (const float*)d_in[28];
  const float* t1b  = (const float*)d_in[29];
  const float* t2w  = (const float*)d_in[30];
  const float* t2b  = (const float*)d_in[31];
  float* out = (float*)d_out;

  // workspace layout (floats)
  float* ws = (float*)d_ws;
  float* bx    = ws;                               // TOK*DM
  float* bxr   = bx    + (size_t)TOK * DM;         // TOK*DM
  float* bh1   = bxr   + (size_t)TOK * DM;         // TOK*DM
  float* bh2   = bh1   + (size_t)TOK * DM;         // TOK*DM
  float* bxz   = bh2   + (size_t)TOK * DM;         // TOK*1024
  float* bxc   = bxz   + (size_t)TOK * 1024;       // TOK*DI
  float* bxdbl = bxc   + (size_t)TOK * DI;         // TOK*48
  float* bdt   = bxdbl + (size_t)TOK * XDW;        // TOK*DI (yv aliased)
  float* blgf  = bdt   + (size_t)TOK * DI;         // TOK
  float* blgb  = blgf  + TOK;                      // TOK
  float* bzp   = blgb  + TOK;                      // NBATCH*512
  float* bzl   = bzp   + NBATCH * 512;             // NBATCH*512

  auto run_block = [&](const float* in, float* outp, const Blk& p) {
    gemm_wmma_f32<<<dim3(1024 / BN, TOK / BM), 256, 0, stream>>>(
        in, DM, p.in_w, bxz, TOK, 1024, DM, nullptr, 0);
    conv_silu<<<(TOK * DI) / 256, 256, 0, stream>>>(bxz, p.conv_w, p.conv_b, bxc);
    gemm_wmma_f32<<<dim3(1, TOK / BM), 256, 0, stream>>>(
        bxc, DI, p.x_w, bxdbl, TOK, XDW, DI, nullptr, 0);
    gemm_wmma_f32<<<dim3(DI / BN, TOK / BM), 256, 0, stream>>>(
        bxdbl, XDW, p.dt_w, bdt, TOK, DI, 16, p.dt_b, 1);
    scan_fused<<<NBATCH * 2, 256, 0, stream>>>(bdt, bxc, bxz, bxdbl, p.A_log, p.D, bdt);
    gemm_wmma_f32<<<dim3(DM / BN, TOK / BM), 256, 0, stream>>>(
        bdt, DI, p.out_w, outp, TOK, DM, DI, nullptr, 0);
  };

  inproj_flip<<<(TOK * DM) / 256, 256, 0, stream>>>(y, ipw, ipb, bx, bxr);

  // forward chain: x -> bh1 -> bh2 ; LN -> bh1
  run_block(bx,  bh1, blk[0]);
  run_block(bh1, bh2, blk[1]);
  layernorm<<<TOK / 8, 256, 0, stream>>>(bh2, ln_g, ln_b, bh1);

  // backward chain: xr -> bx -> bxr ; LN -> bh2
  run_block(bxr, bx,  blk[0]);
  run_block(bx,  bxr, blk[1]);
  layernorm<<<TOK / 8, 256, 0, stream>>>(bxr, ln_g, ln_b, bh2);

  // pooling + last-token
  attn_logits<<<TOK / 8, 256, 0, stream>>>(bh1, attw, blgf);
  attn_logits<<<TOK / 8, 256, 0, stream>>>(bh2, attw, blgb);
  pool_softmax<<<NBATCH, 256, 0, stream>>>(bh1, blgf, bzp, bzl, 0);
  pool_softmax<<<NBATCH, 256, 0, stream>>>(bh2, blgb, bzp, bzl, DM);

  // heads: out_g at [0..511], out_t at [512..1023]
  head_mlp<<<NBATCH, 512, 0, stream>>>(bzp, g1w, g1b, g2w, g2b, out);
  head_mlp<<<NBATCH, 512, 0, stream>>>(bzl, t1w, t1b, t2w, t2b, out + NBATCH * 128);
}